// TreeSelfAttentionGPU_30116310680229
// MI455X (gfx1250) — compile-verified
//
#include <hip/hip_runtime.h>
#include <math.h>

typedef __attribute__((ext_vector_type(2))) float v2f;
typedef __attribute__((ext_vector_type(8))) float v8f;

#define D_DIM 1024
#define T_DIM 4096
#define B_DIM 8
#define NUM_NODES 15
#define KSPLIT 8
#define KCHUNK 256   // 2048 / KSPLIT

// ---------------------------------------------------------------------------
// 1) Leaves: nstate[b, 7+l, d] = mean over 256 timesteps of x[b, l*256+t, d]
// ---------------------------------------------------------------------------
__global__ void tree_leaves_kernel(const float* __restrict__ x,
                                   float* __restrict__ nstate) {
  int bl = blockIdx.x;            // 0..63 : (b, leaf)
  int b  = bl >> 3;
  int l  = bl & 7;
  int d  = blockIdx.y * 256 + threadIdx.x;
  const float* base = x + ((size_t)b * T_DIM + (size_t)l * 256) * D_DIM + d;
  float acc = 0.f;
#pragma unroll 8
  for (int t = 0; t < 256; ++t) acc += base[(size_t)t * D_DIM];
  nstate[((size_t)(b * NUM_NODES + 7 + l)) * D_DIM + d] = acc * (1.0f / 256.0f);
}

// ---------------------------------------------------------------------------
// 2) Data-independent LIF gate chain: spike_out[level][d], 3 levels.
// ---------------------------------------------------------------------------
__global__ void tree_spike_kernel(const float* __restrict__ thr,
                                  float* __restrict__ sp) {
  int d = blockIdx.x * blockDim.x + threadIdx.x;
  if (d >= D_DIM) return;
  float t = thr[d];
  float v = 0.f;
#pragma unroll
  for (int lvl = 0; lvl < 3; ++lvl) {
    float vn    = 0.95f * v;                       // TAU_MEM
    float spike = (vn >= t) ? 1.f : 0.f;
    float va    = vn - spike * t;
    float sur   = 1.f / (1.f + __expf(-25.f * (va - t)));
    sp[lvl * D_DIM + d] = sur * spike;
    v = va;
  }
}

// ---------------------------------------------------------------------------
// 3a) One tree level, K-split WMMA GEMM.
//     fused[b,n,:] = [nstate[b,2n+1,:], nstate[b,2n+2,:]]  (K = 2048)
//     Each wave: one 16x16 tile, one K chunk of 256 (64 WMMA steps).
//     partial[ks][m][d] written; no masks/branches in the hot loop
//     (padded A rows m>=M read a clamped valid row; their D rows are
//      never stored, and WMMA rows are independent dot products).
// ---------------------------------------------------------------------------
__global__ void tree_level_wmma_kernel(const float* __restrict__ nstate,
                                       const float* __restrict__ W,   // (D, 2D) row-major
                                       float* __restrict__ partial,   // [KSPLIT][M][D]
                                       int ls, int n_nodes, int M, int n_tiles_m) {
  const int n_tiles_n = D_DIM / 16;                 // 64
  int wave = (int)((blockIdx.x * blockDim.x + threadIdx.x) >> 5);
  int lane = threadIdx.x & 31;
  int total = n_tiles_m * n_tiles_n * KSPLIT;
  if (wave >= total) return;                        // wave-uniform exit

  int tn = wave % n_tiles_n;
  int tm = (wave / n_tiles_n) % n_tiles_m;
  int ks = wave / (n_tiles_n * n_tiles_m);

  int half = lane >> 4;                             // 0: lanes 0-15, 1: lanes 16-31
  int l16  = lane & 15;

  // A-matrix row (clamped to a valid row; garbage rows never stored)
  int m      = tm * 16 + l16;
  int mm     = (m < M) ? m : 0;
  int b      = mm / n_nodes;
  int nl     = mm % n_nodes;
  int node_l = 2 * (ls + nl) + 1;                   // left child; right child row is +D
  const float* arow = nstate + ((size_t)(b * NUM_NODES + node_l)) * D_DIM;

  // B-matrix column: B[k][n] = W_proj[n][k]
  int n = tn * 16 + l16;
  const float* wrow = W + (size_t)n * (2 * D_DIM);

  v8f c = {0.f, 0.f, 0.f, 0.f, 0.f, 0.f, 0.f, 0.f};
  int kbeg = ks * KCHUNK;
#pragma unroll 8
  for (int k0 = kbeg; k0 < kbeg + KCHUNK; k0 += 4) {
    // 16x4 f32 A frag: VGPR v holds K=k0+v (lanes 0-15) / K=k0+v+2 (lanes 16-31)
    const float* asrc = arow + k0 + 2 * half;
    v2f a;  a[0]  = asrc[0]; a[1]  = asrc[1];
    const float* bsrc = wrow + k0 + 2 * half;
    v2f bb; bb[0] = bsrc[0]; bb[1] = bsrc[1];
    c = __builtin_amdgcn_wmma_f32_16x16x4_f32(false, a, false, bb,
                                              (short)0, c, false, false);
  }

#pragma unroll
  for (int r = 0; r < 8; ++r) {
    int mo = tm * 16 + r + 8 * half;                // D layout: VGPR r -> M=r (+8 hi lanes)
    if (mo < M)
      partial[((size_t)(ks * M + mo)) * D_DIM + n] = c[r];
  }
}

// ---------------------------------------------------------------------------
// 3b) Combine K-split partials, apply gate scale, scatter to parent nodes.
//     Deterministic (fixed summation order), writes every (m,d) of the level.
// ---------------------------------------------------------------------------
__global__ void tree_level_combine_kernel(const float* __restrict__ partial,
                                          const float* __restrict__ sp_level,
                                          float* __restrict__ nstate,
                                          int ls, int n_nodes, int M) {
  int idx = blockIdx.x * 256 + threadIdx.x;         // m*D + d
  if (idx >= M * D_DIM) return;
  int m = idx >> 10;
  int d = idx & (D_DIM - 1);
  float acc = 0.f;
#pragma unroll
  for (int ks = 0; ks < KSPLIT; ++ks)
    acc += partial[((size_t)(ks * M + m)) * D_DIM + d];
  int b  = m / n_nodes;
  int no = ls + (m % n_nodes);
  nstate[((size_t)(b * NUM_NODES + no)) * D_DIM + d] = acc * sp_level[d];
}

// ---------------------------------------------------------------------------
// 4) Per-channel softmax over 15 node weights + mixture[b,d]
// ---------------------------------------------------------------------------
__global__ void tree_mixture_kernel(const float* __restrict__ nw,
                                    const float* __restrict__ nstate,
                                    float* __restrict__ mix) {
  int d = blockIdx.x * 256 + threadIdx.x;
  if (d >= D_DIM) return;
  float w[NUM_NODES];
  float mx = -3.402823466e+38f;
#pragma unroll
  for (int nn = 0; nn < NUM_NODES; ++nn) {
    w[nn] = nw[nn * D_DIM + d];
    mx = fmaxf(mx, w[nn]);
  }
  float ssum = 0.f;
#pragma unroll
  for (int nn = 0; nn < NUM_NODES; ++nn) { w[nn] = __expf(w[nn] - mx); ssum += w[nn]; }
  float inv = 1.f / ssum;
#pragma unroll
  for (int bb = 0; bb < B_DIM; ++bb) {
    float acc = 0.f;
#pragma unroll
    for (int nn = 0; nn < NUM_NODES; ++nn)
      acc += w[nn] * nstate[((size_t)(bb * NUM_NODES + nn)) * D_DIM + d];
    mix[bb * D_DIM + d] = acc * inv;
  }
}

// ---------------------------------------------------------------------------
// 5) out[b,t,:] = LayerNorm(x[b,t,:] + mixture[b,:]) * gamma + beta
//    One block (256 threads = 8 wave32, float4) per row. Wave shuffle reduce.
// ---------------------------------------------------------------------------
__global__ void tree_ln_kernel(const float* __restrict__ x,
                               const float* __restrict__ mix,
                               const float* __restrict__ gamma,
                               const float* __restrict__ beta,
                               float* __restrict__ out) {
  int row = blockIdx.x;                 // b*T + t
  int b   = row >> 12;                  // T = 4096
  int tid = threadIdx.x;

  float4 xv = ((const float4*)(x + (size_t)row * D_DIM))[tid];
  float4 mv = ((const float4*)(mix + (size_t)b * D_DIM))[tid];
  float v0 = xv.x + mv.x, v1 = xv.y + mv.y, v2 = xv.z + mv.z, v3 = xv.w + mv.w;
  float s  = v0 + v1 + v2 + v3;
  float sq = v0 * v0 + v1 * v1 + v2 * v2 + v3 * v3;

  // wave32 shuffle reduce
#pragma unroll
  for (int off = 16; off > 0; off >>= 1) {
    s  += __shfl_xor(s,  off, 32);
    sq += __shfl_xor(sq, off, 32);
  }
  __shared__ float ws0[8];
  __shared__ float ws1[8];
  int wid = tid >> 5, lid = tid & 31;
  if (lid == 0) { ws0[wid] = s; ws1[wid] = sq; }
  __syncthreads();
  if (wid == 0) {
    float a  = (lid < 8) ? ws0[lid] : 0.f;
    float b2 = (lid < 8) ? ws1[lid] : 0.f;
#pragma unroll
    for (int off = 4; off > 0; off >>= 1) {
      a  += __shfl_xor(a,  off, 32);
      b2 += __shfl_xor(b2, off, 32);
    }
    if (lid == 0) { ws0[0] = a; ws1[0] = b2; }
  }
  __syncthreads();
  float mu   = ws0[0] * (1.0f / D_DIM);
  float var  = ws1[0] * (1.0f / D_DIM) - mu * mu;
  float rstd = rsqrtf(var + 1e-5f);

  float4 g  = ((const float4*)gamma)[tid];
  float4 be = ((const float4*)beta)[tid];
  float4 o;
  o.x = (v0 - mu) * rstd * g.x + be.x;
  o.y = (v1 - mu) * rstd * g.y + be.y;
  o.z = (v2 - mu) * rstd * g.z + be.z;
  o.w = (v3 - mu) * rstd * g.w + be.w;
  ((float4*)(out + (size_t)row * D_DIM))[tid] = o;
}

// ---------------------------------------------------------------------------
extern "C" void kernel_launch(void* const* d_in, const int* in_sizes, int n_in,
                              void* d_out, int out_size, void* d_ws, size_t ws_size,
                              hipStream_t stream) {
  const float* x       = (const float*)d_in[0];   // (B,T,D)
  const float* W_proj  = (const float*)d_in[1];   // (D, 2D)
  const float* thresh  = (const float*)d_in[2];   // (D,)
  const float* node_w  = (const float*)d_in[3];   // (15, D)
  const float* gamma   = (const float*)d_in[4];   // (D,)
  const float* beta    = (const float*)d_in[5];   // (D,)
  float* out           = (float*)d_out;           // (B,T,D)

  float* ws      = (float*)d_ws;
  float* nstate  = ws;                                          // B*15*D
  float* spike   = nstate + (size_t)B_DIM * NUM_NODES * D_DIM;  // 3*D
  float* mixture = spike + 3 * D_DIM;                           // B*D
  float* partial = mixture + (size_t)B_DIM * D_DIM;             // KSPLIT*32*D (reused/level)

  // 1) leaves -> nodes 7..14
  tree_leaves_kernel<<<dim3(B_DIM * 8, 4), 256, 0, stream>>>(x, nstate);

  // 2) data-independent gate chain
  tree_spike_kernel<<<dim3(4), 256, 0, stream>>>(thresh, spike);

  // 3) tree levels (level 2 -> 1 -> 0): K-split WMMA + deterministic combine
  //    level 2: ls=3, 4 nodes, M=32 -> 2*64*8 = 1024 waves
  tree_level_wmma_kernel<<<dim3(128), 256, 0, stream>>>(nstate, W_proj, partial, 3, 4, 32, 2);
  tree_level_combine_kernel<<<dim3((32 * D_DIM) / 256), 256, 0, stream>>>(partial, spike + 0 * D_DIM, nstate, 3, 4, 32);
  //    level 1: ls=1, 2 nodes, M=16 -> 512 waves
  tree_level_wmma_kernel<<<dim3(64), 256, 0, stream>>>(nstate, W_proj, partial, 1, 2, 16, 1);
  tree_level_combine_kernel<<<dim3((16 * D_DIM) / 256), 256, 0, stream>>>(partial, spike + 1 * D_DIM, nstate, 1, 2, 16);
  //    level 0: ls=0, 1 node,  M=8 -> 512 waves
  tree_level_wmma_kernel<<<dim3(64), 256, 0, stream>>>(nstate, W_proj, partial, 0, 1, 8, 1);
  tree_level_combine_kernel<<<dim3((8 * D_DIM) / 256), 256, 0, stream>>>(partial, spike + 2 * D_DIM, nstate, 0, 1, 8);

  // 4) softmax-weighted mixture over all 15 nodes
  tree_mixture_kernel<<<dim3(4), 256, 0, stream>>>(node_w, nstate, mixture);

  // 5) fused broadcast-add + LayerNorm (the bandwidth-bound pass)
  tree_ln_kernel<<<dim3(B_DIM * T_DIM), 256, 0, stream>>>(x, mixture, gamma, beta, out);
}